// GraphSAGEEncoderMeanPool_81011673137268
// MI455X (gfx1250) — compile-verified
//
#include <hip/hip_runtime.h>

#define N_NODES 50000
#define N_EDGES 800000
#define D_IN 128
#define D_H 128
#define D_OUT 64

typedef __attribute__((ext_vector_type(2))) float v2f;
typedef __attribute__((ext_vector_type(8))) float v8f;

// ---------------------------------------------------------------- zero fill
__global__ void zero_f32(float* __restrict__ p, long n) {
  long i = (long)blockIdx.x * blockDim.x + threadIdx.x;
  long stride = (long)gridDim.x * blockDim.x;
  for (; i < n; i += stride) p[i] = 0.0f;
}

// --------------------------------------------- edge gather + atomic scatter
// One wave32 per edge; each lane moves 4 consecutive floats (4*32 = 128).
// Gathers are L2-resident (25.6MB node table << 192MB L2); scatter uses
// global_atomic_add_f32 which resolves at L2.
__global__ __launch_bounds__(256) void edge_mean_agg(
    const long long* __restrict__ src, const long long* __restrict__ dst,
    const float* __restrict__ x, float* __restrict__ agg,
    float* __restrict__ cnt, int nEdges, int doCnt) {
  int t = blockIdx.x * blockDim.x + threadIdx.x;
  int e = t >> 5;
  int lane = t & 31;
  if (e >= nEdges) return;
  int s = (int)src[e];
  int d = (int)dst[e];
  const float4 v = *reinterpret_cast<const float4*>(x + (size_t)s * 128 + lane * 4);
  float* a = agg + (size_t)d * 128 + lane * 4;
  atomicAdd(a + 0, v.x);
  atomicAdd(a + 1, v.y);
  atomicAdd(a + 2, v.z);
  atomicAdd(a + 3, v.w);
  if (doCnt && lane == 0) atomicAdd(cnt + d, 1.0f);
}

// ------------------------------------------------------- fused SAGE linear
// out[tile 16 rows] = relu?( (agg/max(cnt,1)) @ Wl + b + x @ Wr )
// One wave per 16-row node tile. f32 WMMA 16x16x4, K = D_IN in steps of 4,
// two K passes (mean side with Wl, root side with Wr) accumulate into the
// same C registers; bias is splatted into C before the K loops.
template <int NT, bool RELU>
__global__ __launch_bounds__(128) void sage_linear(
    const float* __restrict__ agg, const float* __restrict__ cnt,
    const float* __restrict__ xr, const float* __restrict__ Wl,
    const float* __restrict__ bias, const float* __restrict__ Wr,
    float* __restrict__ out) {
  constexpr int dOut = NT * 16;
  const int lane = threadIdx.x & 31;
  const int wave = threadIdx.x >> 5;
  const int tile = blockIdx.x * 4 + wave;
  if (tile * 16 >= N_NODES) return;  // wave-uniform exit -> EXEC all-ones for WMMA

  const int rowl = lane & 15;          // A row / B,C column within tile
  const int hi   = lane >> 4;          // half-wave select
  const int koff = hi * 2;             // A holds K={koff, koff+1}; B VGPR v -> K = koff + v
  const int row  = tile * 16 + rowl;

  const float c   = cnt[row];
  const float rcp = 1.0f / fmaxf(c, 1.0f);

  v8f acc[NT];
#pragma unroll
  for (int nt = 0; nt < NT; ++nt) {
    const float bv = bias[nt * 16 + rowl];
    v8f t = {bv, bv, bv, bv, bv, bv, bv, bv};
    acc[nt] = t;
  }

  const float* aRow = agg + (size_t)row * D_IN;
  const float* xRow = xr + (size_t)row * D_IN;

  // --- pass 1: mean-aggregated neighbors @ Wl ---
#pragma unroll 2
  for (int k = 0; k < D_IN; k += 4) {
    const float2 av = *reinterpret_cast<const float2*>(aRow + k + koff);
    v2f a;
    a.x = av.x * rcp;
    a.y = av.y * rcp;
#pragma unroll
    for (int nt = 0; nt < NT; ++nt) {
      v2f b;
      b.x = Wl[(size_t)(k + koff) * dOut + nt * 16 + rowl];
      b.y = Wl[(size_t)(k + koff + 1) * dOut + nt * 16 + rowl];
      acc[nt] = __builtin_amdgcn_wmma_f32_16x16x4_f32(
          false, a, false, b, (short)0, acc[nt], false, false);
    }
  }

  // --- pass 2: root features @ Wr ---
#pragma unroll 2
  for (int k = 0; k < D_IN; k += 4) {
    const float2 xv = *reinterpret_cast<const float2*>(xRow + k + koff);
    v2f a;
    a.x = xv.x;
    a.y = xv.y;
#pragma unroll
    for (int nt = 0; nt < NT; ++nt) {
      v2f b;
      b.x = Wr[(size_t)(k + koff) * dOut + nt * 16 + rowl];
      b.y = Wr[(size_t)(k + koff + 1) * dOut + nt * 16 + rowl];
      acc[nt] = __builtin_amdgcn_wmma_f32_16x16x4_f32(
          false, a, false, b, (short)0, acc[nt], false, false);
    }
  }

  // --- epilogue: C layout M = v + 8*hi, N = nt*16 + rowl ---
  const int mbase = tile * 16 + hi * 8;
#pragma unroll
  for (int nt = 0; nt < NT; ++nt) {
#pragma unroll
    for (int v = 0; v < 8; ++v) {
      float val = acc[nt][v];
      if (RELU) val = fmaxf(val, 0.0f);
      out[(size_t)(mbase + v) * dOut + nt * 16 + rowl] = val;
    }
  }
}

// ---------------------------------------------------------------- launcher
extern "C" void kernel_launch(void* const* d_in, const int* in_sizes, int n_in,
                              void* d_out, int out_size, void* d_ws, size_t ws_size,
                              hipStream_t stream) {
  (void)in_sizes; (void)n_in; (void)out_size; (void)ws_size;

  const float*     x   = (const float*)d_in[0];
  const long long* ei  = (const long long*)d_in[1];  // int64 edge_index [2, E]
  const float*     W1l = (const float*)d_in[2];
  const float*     b1  = (const float*)d_in[3];
  const float*     W1r = (const float*)d_in[4];
  const float*     W2l = (const float*)d_in[5];
  const float*     b2  = (const float*)d_in[6];
  const float*     W2r = (const float*)d_in[7];
  float*           out = (float*)d_out;

  const long long* srcI = ei;
  const long long* dstI = ei + N_EDGES;

  // Workspace layout: cnt | agg | h  (cnt and agg contiguous for one zero pass)
  float* cnt = (float*)d_ws;                    // N_NODES
  float* agg = cnt + N_NODES;                   // N_NODES * 128 (16B aligned: 200000 % 16 == 0)
  float* h   = agg + (size_t)N_NODES * 128;     // N_NODES * 128

  const int edgeBlocks = (N_EDGES * 32) / 256;            // 100000, exact
  const int tileBlocks = (N_NODES / 16 + 3) / 4;          // 782 blocks of 4 waves

  // ---- layer 1 ----
  zero_f32<<<2048, 256, 0, stream>>>(cnt, (long)N_NODES * (1 + 128));
  edge_mean_agg<<<edgeBlocks, 256, 0, stream>>>(srcI, dstI, x, agg, cnt, N_EDGES, 1);
  sage_linear<8, true><<<tileBlocks, 128, 0, stream>>>(agg, cnt, x, W1l, b1, W1r, h);

  // ---- layer 2 (reuse cnt; degrees are identical) ----
  zero_f32<<<2048, 256, 0, stream>>>(agg, (long)N_NODES * 128);
  edge_mean_agg<<<edgeBlocks, 256, 0, stream>>>(srcI, dstI, h, agg, cnt, N_EDGES, 0);
  sage_linear<4, false><<<tileBlocks, 128, 0, stream>>>(agg, cnt, h, W2l, b2, W2r, out);
}